// AdaptiveSpline_38293928411786
// MI455X (gfx1250) — compile-verified
//
#include <hip/hip_runtime.h>

#define NUM_KNOTS 64
#define DEGREE    3
#define KBASIS    60  // NUM_KNOTS - 1 - DEGREE

typedef __attribute__((ext_vector_type(2))) float v2f;
typedef __attribute__((ext_vector_type(8))) float v8f;

// Per-point spline evaluation using LDS tables.
// kn: 64 knots, cf: coeffs (zero-padded to 64), invd: 60x3 reciprocal denoms,
// pref: inclusive prefix sums of cf.
__device__ __forceinline__ float eval_point(float xp, float k0, float scale,
                                            const float* kn, const float* cf,
                                            const float* invd, const float* pref) {
    // Locate knot interval m with kn[m] <= xp < kn[m+1] (uniform estimate + fixup).
    int m = (int)((xp - k0) * scale);
    m = min(max(m, 0), NUM_KNOTS - 1);
    float kl = kn[m];                               // unconditional LDS loads
    float kr = kn[min(m + 1, NUM_KNOTS - 1)];
    if (xp < kl)                                  m = max(m - 1, 0);
    else if (m < NUM_KNOTS - 1 && xp >= kr)       m = m + 1;

    // All basis j <= m-3 are exactly 1 (x >= t_{j+3}); all j >= m+1 are 0.
    int   j0 = m - 3;
    float pv = pref[min(max(j0, 0), KBASIS - 1)];   // unconditional load
    float acc = (j0 >= 0) ? pv : 0.0f;              // v_cndmask on the value

    // Up to 3 fractional terms, evaluated with the exact reference formula.
#pragma unroll
    for (int t = 0; t < 3; ++t) {
        int   j  = m - 2 + t;
        int   jc = min(max(j, 0), KBASIS - 1);      // clamp for safe LDS access
        float dx = xp - kn[jc];
        float r0 = __builtin_amdgcn_fmed3f(dx * invd[jc * 3 + 0], 0.0f, 1.0f);
        float r1 = __builtin_amdgcn_fmed3f(dx * invd[jc * 3 + 1], 0.0f, 1.0f);
        float r2 = __builtin_amdgcn_fmed3f(dx * invd[jc * 3 + 2], 0.0f, 1.0f);
        float cv = cf[jc];                          // unconditional load
        float cj = (j == jc) ? cv : 0.0f;           // zero if j out of range
        acc = fmaf(r0 * r1 * r2, cj, acc);
    }
    return acc;
}

__global__ __launch_bounds__(256) void AdaptiveSpline_38293928411786_kernel(
    const float* __restrict__ x, const float* __restrict__ knots,
    const float* __restrict__ coeffs, float* __restrict__ out, int n) {
    __shared__ float kn[NUM_KNOTS];
    __shared__ float cf[NUM_KNOTS];        // zero padded past KBASIS
    __shared__ float invd[KBASIS * 3];
    __shared__ float pref[NUM_KNOTS];      // pref[j] = sum_{i<=j} cf[i]

    const int tid = threadIdx.x;
    if (tid < NUM_KNOTS) {
        kn[tid] = knots[tid];
        cf[tid] = (tid < KBASIS) ? coeffs[tid] : 0.0f;
    }
    __syncthreads();

    // Reciprocal denominators (reference: knots[j+i] - knots[j]).
    if (tid < KBASIS) {
        float t0 = kn[tid];
        invd[tid * 3 + 0] = 1.0f / (kn[tid + 1] - t0);
        invd[tid * 3 + 1] = 1.0f / (kn[tid + 2] - t0);
        invd[tid * 3 + 2] = 1.0f / (kn[tid + 3] - t0);
    }

    // Wave 0: prefix sums of coefficients via WMMA f32 16x16x4.
    // P = L * c, L = 64x64 lower-triangular ones. Masked coeffs go in A
    // (documented 16x4 f32 layout); B is all-ones so its layout is irrelevant
    // (every column of D equals the prefix chunk).
    if (tid < 32) {
        const int lane  = tid;
        const int mrow  = lane & 15;   // M = lane % 16
        const int khalf = lane >> 4;   // lanes 16..31 hold K = 2,3
        v2f b;
        b[0] = 1.0f;
        b[1] = 1.0f;
        for (int mt = 0; mt < 4; ++mt) {
            v8f d = {0.0f, 0.0f, 0.0f, 0.0f, 0.0f, 0.0f, 0.0f, 0.0f};
            const int mg = mt * 16 + mrow;
            for (int kt = 0; kt < 16; ++kt) {
                const int k0i = kt * 4 + khalf * 2;   // always in [0, 63]
                float c0 = cf[k0i];                   // unconditional DS loads
                float c1 = cf[k0i + 1];
                v2f a;
                a[0] = (k0i     <= mg) ? c0 : 0.0f;   // VGPR0: K = k0i
                a[1] = (k0i + 1 <= mg) ? c1 : 0.0f;   // VGPR1: K = k0i+1
                d = __builtin_amdgcn_wmma_f32_16x16x4_f32(
                        false, a, false, b, (short)0, d, false, false);
            }
            if (mrow == 0) {  // lane 0 -> M = r, lane 16 -> M = r + 8
                const int mbase = mt * 16 + khalf * 8;
#pragma unroll
                for (int r = 0; r < 8; ++r) pref[mbase + r] = d[r];
            }
        }
    }
    __syncthreads();

    const float k0    = kn[0];
    const float scale = 63.0f / (kn[NUM_KNOTS - 1] - k0);

    const int nvec    = n >> 2;
    const int gstride = gridDim.x * blockDim.x;
    const int gid     = blockIdx.x * blockDim.x + tid;

    const float4* __restrict__ x4 = (const float4*)x;
    float4* __restrict__ o4       = (float4*)out;

    for (int i = gid; i < nvec; i += gstride) {
        float4 xv = x4[i];
        float4 ov;
        ov.x = eval_point(xv.x, k0, scale, kn, cf, invd, pref);
        ov.y = eval_point(xv.y, k0, scale, kn, cf, invd, pref);
        ov.z = eval_point(xv.z, k0, scale, kn, cf, invd, pref);
        ov.w = eval_point(xv.w, k0, scale, kn, cf, invd, pref);
        o4[i] = ov;
    }

    // Scalar tail (N is a multiple of 4 here, but stay robust).
    for (int i = (nvec << 2) + gid; i < n; i += gstride) {
        out[i] = eval_point(x[i], k0, scale, kn, cf, invd, pref);
    }
}

extern "C" void kernel_launch(void* const* d_in, const int* in_sizes, int n_in,
                              void* d_out, int out_size, void* d_ws, size_t ws_size,
                              hipStream_t stream) {
    const float* x      = (const float*)d_in[0];
    const float* knots  = (const float*)d_in[1];
    const float* coeffs = (const float*)d_in[2];
    float* out          = (float*)d_out;
    const int n         = in_sizes[0];

    const int threads = 256;            // 8 wave32 per block
    int nvec   = (n + 3) >> 2;
    int blocks = (nvec + threads - 1) / threads;   // 2048 for N = 2M
    if (blocks < 1) blocks = 1;
    if (blocks > 65535) blocks = 65535;

    AdaptiveSpline_38293928411786_kernel<<<blocks, threads, 0, stream>>>(
        x, knots, coeffs, out, n);
}